// UniformBertSelfAttention_14448269983908
// MI455X (gfx1250) — compile-verified
//
#include <hip/hip_runtime.h>
#include <hip/hip_bf16.h>
#include <math.h>

// Problem constants (match reference)
#define B_ 4
#define L_ 4096
#define D_ 768
#define H_ 12
#define KT_ 32                 // k-split in stage B
#define KCHUNK (L_ / KT_)      // 128

typedef __attribute__((ext_vector_type(2))) float v2f;
typedef __attribute__((ext_vector_type(8))) float v8f;
typedef __attribute__((ext_vector_type(4))) float f4v;

// ---------------------------------------------------------------------------
// Stage A: per-batch softmax weights over mask, inverse sum, zero accumulator.
// grid = B_, block = 256
// ---------------------------------------------------------------------------
__global__ void stageA_softmax(const float* __restrict__ mask,
                               float* __restrict__ w,
                               float* __restrict__ invsum,
                               float* __restrict__ hsum) {
  __shared__ float red[256];
  const int b = blockIdx.x;
  const int t = threadIdx.x;
  const float* mrow = mask + (size_t)b * L_;

  // max over L
  float m = -INFINITY;
  for (int k = t; k < L_; k += 256) m = fmaxf(m, mrow[k]);
  red[t] = m; __syncthreads();
  for (int s = 128; s > 0; s >>= 1) {
    if (t < s) red[t] = fmaxf(red[t], red[t + s]);
    __syncthreads();
  }
  m = red[0];
  __syncthreads();

  // exp(mask - m), write weights, sum
  float ssum = 0.f;
  for (int k = t; k < L_; k += 256) {
    float e = expf(mrow[k] - m);
    w[(size_t)b * L_ + k] = e;
    ssum += e;
  }
  red[t] = ssum; __syncthreads();
  for (int s = 128; s > 0; s >>= 1) {
    if (t < s) red[t] += red[t + s];
    __syncthreads();
  }
  if (t == 0) invsum[b] = 1.0f / red[0];

  // zero the atomic accumulator for this batch
  for (int d = t; d < D_; d += 256) hsum[(size_t)b * D_ + d] = 0.f;
}

// ---------------------------------------------------------------------------
// Stage B: hsum[b,d] += sum_{k in tile} w[b,k] * hidden[b,k,d]
// grid = (D_/256, KT_, B_), block = 256.  This is the 48 MB read kernel.
// ---------------------------------------------------------------------------
__global__ void stageB_weighted_sum(const float* __restrict__ hidden,
                                    const float* __restrict__ w,
                                    float* __restrict__ hsum) {
  const int d  = blockIdx.x * 256 + threadIdx.x;   // 0..767
  const int k0 = blockIdx.y * KCHUNK;
  const int b  = blockIdx.z;

  const float* __restrict__ hp = hidden + ((size_t)b * L_ + k0) * D_ + d;
  const float* __restrict__ wp = w + (size_t)b * L_ + k0;

  float acc = 0.f;
#pragma unroll 4
  for (int k = 0; k < KCHUNK; ++k) {
    acc = fmaf(wp[k], hp[(size_t)k * D_], acc);
  }
  atomicAdd(&hsum[(size_t)b * D_ + d], acc);
}

// ---------------------------------------------------------------------------
// Stage C: ctx[m, n] = sum_k (hsum[m,k]*invsum[m]) * Wv[n,k] + bv[n]
// fp32 WMMA: D(16x16) = A(16x4) x B(4x16) + C, chained over K=768.
// One wave (32 threads) per 16-wide N tile; grid = 48, block = 32.
// A layout (ISA 7.12.2, 32-bit A 16x4):
//   lanes 0-15 : a0=A[m][k0+0], a1=A[m][k0+1]      (m = lane)
//   lanes 16-31: a0=A[m][k0+2], a1=A[m][k0+3]      (m = lane-16)
// B layout mirrors with lane = N.
// C/D layout: c[r] = D[r + 8*half][n0 + lane%16].
// ---------------------------------------------------------------------------
__global__ void stageC_wmma_gemm(const float* __restrict__ hsum,
                                 const float* __restrict__ invsum,
                                 const float* __restrict__ Wv,
                                 const float* __restrict__ bv,
                                 float* __restrict__ ctx) {
  const int lane   = threadIdx.x;       // 0..31
  const int half   = lane >> 4;         // 0: K+0/1, 1: K+2/3
  const int lane16 = lane & 15;
  const int n0     = blockIdx.x * 16;

  const int  m      = lane16;                       // A row this lane carries
  const int  mclamp = (m < B_) ? m : 0;             // keep loads in-bounds
  const float mmask = (m < B_) ? 1.0f : 0.0f;       // zero padding rows
  const float scale = invsum[mclamp] * mmask;

  const float* __restrict__ arow = hsum + (size_t)mclamp * D_;
  const float* __restrict__ brow = Wv + (size_t)(n0 + lane16) * D_; // B[k][n] = Wv[n][k]

  v8f c = {};
#pragma unroll 8
  for (int k0 = 0; k0 < D_; k0 += 4) {
    const int ka = k0 + half * 2;
    v2f a, bb;
    a.x  = arow[ka]     * scale;
    a.y  = arow[ka + 1] * scale;
    bb.x = brow[ka];
    bb.y = brow[ka + 1];
    // 8 args: (neg_a, A, neg_b, B, c_mod, C, reuse_a, reuse_b)
    c = __builtin_amdgcn_wmma_f32_16x16x4_f32(
        false, a, false, bb, (short)0, c, false, false);
  }

  // Write valid rows M=0..3 (held by lanes 0-15, vgprs 0..3) with bias.
  if (half == 0) {
    const float bias = bv[n0 + lane16];
#pragma unroll
    for (int r = 0; r < B_; ++r) {
      ctx[(size_t)r * D_ + n0 + lane16] = c[r] + bias;
    }
  }
}

// ---------------------------------------------------------------------------
// Stage D: broadcast ctx[b,:] to out[b,l,:].  48 MB of b128 stores.
// grid = B_*L_*(D_/4)/256, block = 256.
// ---------------------------------------------------------------------------
__global__ void stageD_broadcast(const float* __restrict__ ctx,
                                 float* __restrict__ out) {
  const size_t i  = (size_t)blockIdx.x * blockDim.x + threadIdx.x; // float4 idx
  const int    D4 = D_ / 4;                                        // 192
  const size_t perB = (size_t)L_ * D4;
  const int b  = (int)(i / perB);
  const int d4 = (int)(i % (size_t)D4);
  const f4v* __restrict__ csrc = (const f4v*)ctx;
  ((f4v*)out)[i] = csrc[(size_t)b * D4 + d4];
}

// ---------------------------------------------------------------------------
extern "C" void kernel_launch(void* const* d_in, const int* in_sizes, int n_in,
                              void* d_out, int out_size, void* d_ws, size_t ws_size,
                              hipStream_t stream) {
  // setup_inputs order: hidden_states, attention_mask, Wq, bq, Wk, bk, Wv, bv
  const float* hidden = (const float*)d_in[0];
  const float* mask   = (const float*)d_in[1];
  const float* Wv     = (const float*)d_in[6];
  const float* bv     = (const float*)d_in[7];
  float* out = (float*)d_out;

  // Workspace carve-up (floats). ~90 KB total.
  float* ws     = (float*)d_ws;
  float* w      = ws;                       // B_*L_   = 16384
  float* invsum = ws + B_ * L_;             // B_      (padded to 16)
  float* hsum   = ws + B_ * L_ + 16;        // B_*D_   = 3072
  float* ctx    = hsum + B_ * D_;           // B_*D_   = 3072

  stageA_softmax   <<<B_, 256, 0, stream>>>(mask, w, invsum, hsum);
  stageB_weighted_sum<<<dim3(D_ / 256, KT_, B_), 256, 0, stream>>>(hidden, w, hsum);
  stageC_wmma_gemm <<<D_ / 16, 32, 0, stream>>>(hsum, invsum, Wv, bv, ctx);
  stageD_broadcast <<<(B_ * L_ * (D_ / 4)) / 256, 256, 0, stream>>>(ctx, out);
}